// CNN_decomp_63952063037876
// MI455X (gfx1250) — compile-verified
//
#include <hip/hip_runtime.h>
#include <hip/hip_bf16.h>
#include <math.h>

// ---------------------------------------------------------------------------
// CP-decomposed CNN on MI455X (gfx1250, wave32).
// Rank/filter contractions use V_WMMA_F32_16X16X4_F32 (exact fp32 — workload
// is bandwidth-bound at 23.3 TB/s, so low-precision WMMA buys nothing).
// Activations kept K-planar so WMMA A-fragment loads coalesce.
// ---------------------------------------------------------------------------

typedef __attribute__((ext_vector_type(2))) float v2f;
typedef __attribute__((ext_vector_type(8))) float v8f;

#define BATCH   512
#define HO1     30          // layer-1 output spatial
#define NPIX1   (BATCH*HO1*HO1)   // 460800  (divisible by 16)
#define HO2     28          // layer-2 output spatial
#define PPB2    (HO2*HO2)         // 784     (divisible by 16)
#define NPIX2   (BATCH*PPB2)      // 401408  (divisible by 16)
#define RANK    16
#define NFILT   32

__device__ __forceinline__ v8f wmma_f32_k4(v2f a, v2f b, v8f c) {
  // 8 args: (neg_a, A, neg_b, B, c_mod, C, reuse_a, reuse_b)
  return __builtin_amdgcn_wmma_f32_16x16x4_f32(false, a, false, b, (short)0, c,
                                               false, false);
}

// --------------------------------------------------------------------------
// Kernel 1: layer-1 separable 3x3x3 CP stencil.
// t1[r*NPIX1 + pix] = sum_{i,j,c} x[b,c,ho+i,wo+j] * f1[i,r]*f2[j,r]*f3[c,r]
// grid: (NPIX1/256, 16), block 256
// --------------------------------------------------------------------------
__global__ __launch_bounds__(256) void layer1_front(
    const float* __restrict__ x, const float* __restrict__ f1,
    const float* __restrict__ f2, const float* __restrict__ f3,
    float* __restrict__ t1) {
  const int pix = blockIdx.x * 256 + threadIdx.x;      // < NPIX1
  const int r   = blockIdx.y;                          // < RANK
  const int b   = pix / 900;
  const int rem = pix - b * 900;
  const int ho  = rem / 30;
  const int wo  = rem - ho * 30;
  const float w3_0 = f3[r], w3_1 = f3[16 + r], w3_2 = f3[32 + r];
  float acc = 0.f;
#pragma unroll
  for (int i = 0; i < 3; ++i) {
    const float w1 = f1[i * 16 + r];
#pragma unroll
    for (int j = 0; j < 3; ++j) {
      const float w12 = w1 * f2[j * 16 + r];
      const size_t base = (((size_t)b * 3) * 32 + (ho + i)) * 32 + (wo + j);
      acc = fmaf(w12 * w3_0, x[base],            acc);
      acc = fmaf(w12 * w3_1, x[base + 32 * 32],  acc);
      acc = fmaf(w12 * w3_2, x[base + 2*32*32],  acc);
    }
  }
  t1[(size_t)r * NPIX1 + pix] = acc;
}

// --------------------------------------------------------------------------
// Kernel 2: layer-1 rank->filter expansion, WMMA f32.
// h1[f*NPIX1 + pix] = sum_r t1[r*NPIX1 + pix] * l1_f0[f,r]
// One wave per 16-pixel x 16-filter tile; 4 waves per block.
// --------------------------------------------------------------------------
__global__ __launch_bounds__(128) void expand1_wmma(
    const float* __restrict__ t1, const float* __restrict__ f0,
    float* __restrict__ h1) {
  const int wave = (blockIdx.x * 128 + threadIdx.x) >> 5;
  const int lane = threadIdx.x & 31;
  const int m = lane & 15;
  const int g = lane >> 4;
  const int pb = (wave >> 1) << 4;   // pixel tile base
  const int fb = (wave & 1) << 4;    // filter tile base
  v8f c = {0.f, 0.f, 0.f, 0.f, 0.f, 0.f, 0.f, 0.f};
#pragma unroll
  for (int kb = 0; kb < RANK; kb += 4) {
    const int k0 = kb + 2 * g;       // per-lane K base (A/B 16x4 layout)
    v2f a, bm;
    a.x  = t1[(size_t)(k0 + 0) * NPIX1 + pb + m];
    a.y  = t1[(size_t)(k0 + 1) * NPIX1 + pb + m];
    bm.x = f0[(fb + m) * RANK + (k0 + 0)];   // B[k][n] = f0[n,k]
    bm.y = f0[(fb + m) * RANK + (k0 + 1)];
    c = wmma_f32_k4(a, bm, c);
  }
#pragma unroll
  for (int j = 0; j < 8; ++j)        // D: vgpr j -> row j (lo) / j+8 (hi)
    h1[(size_t)(fb + m) * NPIX1 + (pb + j + 8 * g)] = c[j];
}

// --------------------------------------------------------------------------
// Kernel 3: layer-2 channel contraction (F=32 -> R=16), WMMA f32, K=32.
// s2[r*NPIX1 + pix] = sum_f h1[f*NPIX1 + pix] * l2_f3[f,r]
// --------------------------------------------------------------------------
__global__ __launch_bounds__(128) void chan2_wmma(
    const float* __restrict__ h1, const float* __restrict__ f3,
    float* __restrict__ s2) {
  const int wave = (blockIdx.x * 128 + threadIdx.x) >> 5;
  const int lane = threadIdx.x & 31;
  const int m = lane & 15;
  const int g = lane >> 4;
  const int pb = wave << 4;          // N=16 -> one filter tile
  v8f c = {0.f, 0.f, 0.f, 0.f, 0.f, 0.f, 0.f, 0.f};
#pragma unroll
  for (int kb = 0; kb < NFILT; kb += 4) {
    const int k0 = kb + 2 * g;
    v2f a, bm;
    a.x  = h1[(size_t)(k0 + 0) * NPIX1 + pb + m];
    a.y  = h1[(size_t)(k0 + 1) * NPIX1 + pb + m];
    bm.x = f3[(k0 + 0) * RANK + m];  // B[k][n] = f3[k,n]
    bm.y = f3[(k0 + 1) * RANK + m];
    c = wmma_f32_k4(a, bm, c);
  }
#pragma unroll
  for (int j = 0; j < 8; ++j)
    s2[(size_t)m * NPIX1 + (pb + j + 8 * g)] = c[j];
}

// --------------------------------------------------------------------------
// Kernel 4: layer-2 separable 3x3 spatial stencil over rank planes.
// t2[r*NPIX2 + pix2] = sum_{i,j} s2[r,b,ho+i,wo+j] * l2_f1[i,r]*l2_f2[j,r]
// grid: (NPIX2/256, 16)
// --------------------------------------------------------------------------
__global__ __launch_bounds__(256) void layer2_spatial(
    const float* __restrict__ s2, const float* __restrict__ f1,
    const float* __restrict__ f2, float* __restrict__ t2) {
  const int pix = blockIdx.x * 256 + threadIdx.x;      // < NPIX2
  const int r   = blockIdx.y;
  const int b   = pix / PPB2;
  const int rem = pix - b * PPB2;
  const int ho  = rem / HO2;
  const int wo  = rem - ho * HO2;
  const float* sp = s2 + (size_t)r * NPIX1 + (size_t)b * 900;
  float acc = 0.f;
#pragma unroll
  for (int i = 0; i < 3; ++i) {
    const float w1 = f1[i * 16 + r];
#pragma unroll
    for (int j = 0; j < 3; ++j)
      acc = fmaf(w1 * f2[j * 16 + r], sp[(ho + i) * 30 + (wo + j)], acc);
  }
  t2[(size_t)r * NPIX2 + pix] = acc;
}

// --------------------------------------------------------------------------
// Kernel 5: layer-2 rank->filter expansion, WMMA f32; writes h2 in the
// reference's channel-major flatten order (b, f, ho, wo). 784 % 16 == 0 so a
// pixel tile never crosses a batch boundary.
// --------------------------------------------------------------------------
__global__ __launch_bounds__(128) void expand2_wmma(
    const float* __restrict__ t2, const float* __restrict__ f0,
    float* __restrict__ h2) {
  const int wave = (blockIdx.x * 128 + threadIdx.x) >> 5;
  const int lane = threadIdx.x & 31;
  const int m = lane & 15;
  const int g = lane >> 4;
  const int pb = (wave >> 1) << 4;
  const int fb = (wave & 1) << 4;
  const int b  = pb / PPB2;                 // uniform per wave
  const int rembase = pb - b * PPB2;
  v8f c = {0.f, 0.f, 0.f, 0.f, 0.f, 0.f, 0.f, 0.f};
#pragma unroll
  for (int kb = 0; kb < RANK; kb += 4) {
    const int k0 = kb + 2 * g;
    v2f a, bm;
    a.x  = t2[(size_t)(k0 + 0) * NPIX2 + pb + m];
    a.y  = t2[(size_t)(k0 + 1) * NPIX2 + pb + m];
    bm.x = f0[(fb + m) * RANK + (k0 + 0)];
    bm.y = f0[(fb + m) * RANK + (k0 + 1)];
    c = wmma_f32_k4(a, bm, c);
  }
  float* outp = h2 + (size_t)b * (NFILT * PPB2) + (size_t)(fb + m) * PPB2;
#pragma unroll
  for (int j = 0; j < 8; ++j)
    outp[rembase + j + 8 * g] = c[j];
}

// --------------------------------------------------------------------------
// Kernel 6: classifier (512 x 25088) @ (25088 x 10) + bias + log_softmax.
// One 256-thread block per batch row; LDS tree reduction over 10 classes.
// --------------------------------------------------------------------------
__global__ __launch_bounds__(256) void classifier_lsm(
    const float* __restrict__ h2, const float* __restrict__ W,
    const float* __restrict__ bias, float* __restrict__ out) {
  const int b = blockIdx.x;
  const int tid = threadIdx.x;
  __shared__ float red[10 * 256];
  float acc[10];
#pragma unroll
  for (int n = 0; n < 10; ++n) acc[n] = 0.f;
  const float* hp = h2 + (size_t)b * 25088;
  for (int i = tid; i < 25088; i += 256) {
    const float v = hp[i];
#pragma unroll
    for (int n = 0; n < 10; ++n) acc[n] = fmaf(v, W[n * 25088 + i], acc[n]);
  }
#pragma unroll
  for (int n = 0; n < 10; ++n) red[n * 256 + tid] = acc[n];
  __syncthreads();
  for (int s = 128; s > 0; s >>= 1) {
    if (tid < s) {
#pragma unroll
      for (int n = 0; n < 10; ++n)
        red[n * 256 + tid] += red[n * 256 + tid + s];
    }
    __syncthreads();
  }
  if (tid < 10) {
    float lg[10];
    float mx = -3.0e38f;
#pragma unroll
    for (int n = 0; n < 10; ++n) {
      lg[n] = red[n * 256] + bias[n];
      mx = fmaxf(mx, lg[n]);
    }
    float se = 0.f;
#pragma unroll
    for (int n = 0; n < 10; ++n) se += expf(lg[n] - mx);
    out[b * 10 + tid] = lg[tid] - mx - logf(se);
  }
}

// --------------------------------------------------------------------------
extern "C" void kernel_launch(void* const* d_in, const int* in_sizes, int n_in,
                              void* d_out, int out_size, void* d_ws,
                              size_t ws_size, hipStream_t stream) {
  (void)in_sizes; (void)n_in; (void)out_size; (void)ws_size;
  const float* x    = (const float*)d_in[0];
  const float* l1f0 = (const float*)d_in[1];
  const float* l1f1 = (const float*)d_in[2];
  const float* l1f2 = (const float*)d_in[3];
  const float* l1f3 = (const float*)d_in[4];
  const float* l2f0 = (const float*)d_in[5];
  const float* l2f1 = (const float*)d_in[6];
  const float* l2f2 = (const float*)d_in[7];
  const float* l2f3 = (const float*)d_in[8];
  const float* Wc   = (const float*)d_in[9];
  const float* bc   = (const float*)d_in[10];
  float* out = (float*)d_out;
  float* ws  = (float*)d_ws;

  // workspace layout (float offsets), with buffer reuse; peak ~118 MB
  float* t1 = ws;                   // 16 * 460800 = 7,372,800
  float* h1 = ws + 8000000;         // 32 * 460800 = 14,745,600
  float* s2 = ws;                   // reuse t1 region (t1 dead)
  float* t2 = ws + 23000000;        // 16 * 401408 = 6,422,528
  float* h2 = ws + 8000000;         // reuse h1 region (h1 dead)

  layer1_front<<<dim3(NPIX1 / 256, RANK), 256, 0, stream>>>(x, l1f1, l1f2,
                                                            l1f3, t1);
  // 28800 pixel tiles * 2 filter tiles = 57600 waves / 4 per block
  expand1_wmma<<<14400, 128, 0, stream>>>(t1, l1f0, h1);
  // 28800 waves / 4 per block
  chan2_wmma<<<7200, 128, 0, stream>>>(h1, l2f3, s2);
  layer2_spatial<<<dim3(NPIX2 / 256, RANK), 256, 0, stream>>>(s2, l2f1, l2f2,
                                                              t2);
  // 25088 pixel tiles * 2 filter tiles = 50176 waves / 4 per block
  expand2_wmma<<<12544, 128, 0, stream>>>(t2, l2f0, h2);
  classifier_lsm<<<BATCH, 256, 0, stream>>>(h2, Wc, bc, out);
}